// SAGE_4123168604186
// MI455X (gfx1250) — compile-verified
//
#include <hip/hip_runtime.h>

// ---------------------------------------------------------------------------
// GraphSAGE 3-layer forward for MI455X (gfx1250, wave32).
//  - Aggregation (bandwidth-bound): edge-parallel float4 gather + f32 atomics.
//  - GEMMs: V_WMMA_F32_16X16X4_F32. A tiles staged through LDS (coalesced
//    global float4 -> bank-conflict-free ds_load_b64 fragments); weights for
//    layer 3 pre-packed to a zero-padded 128x48 so the WMMA loop has no
//    predication at all (EXEC all-ones, no exec save/restore branches).
// ---------------------------------------------------------------------------

#define NN 50000
#define NE 800000
#define FIN 128
#define FH  128
#define FC  47
#define FCP 48   // FC padded to multiple of 16

typedef float v2f __attribute__((ext_vector_type(2)));
typedef float v8f __attribute__((ext_vector_type(8)));

__global__ void zero_f32(float* __restrict__ p, int n) {
  int i = blockIdx.x * blockDim.x + threadIdx.x;
  if (i < n) p[i] = 0.0f;
}

__global__ void degree_kernel(const int* __restrict__ dst, float* __restrict__ deg, int E) {
  int i = blockIdx.x * blockDim.x + threadIdx.x;
  if (i < E) atomicAdd(&deg[dst[i]], 1.0f);
}

__global__ void invdeg_kernel(float* __restrict__ deg, int n) {
  int i = blockIdx.x * blockDim.x + threadIdx.x;
  if (i < n) deg[i] = 1.0f / fmaxf(deg[i], 1.0f);
}

// Zero-pad layer-3 weights/bias: W [128 x 47] -> Wp [128 x 48], b [47] -> bp [48].
__global__ void pack_pad(const float* __restrict__ Ws, const float* __restrict__ Wn,
                         const float* __restrict__ b,
                         float* __restrict__ Wsp, float* __restrict__ Wnp,
                         float* __restrict__ bp) {
  int i = blockIdx.x * blockDim.x + threadIdx.x;
  if (i < FH * FCP) {
    int r = i / FCP, c = i % FCP;
    float vs = (c < FC) ? Ws[r * FC + c] : 0.0f;
    float vn = (c < FC) ? Wn[r * FC + c] : 0.0f;
    Wsp[i] = vs;
    Wnp[i] = vn;
  }
  if (i < FCP) bp[i] = (i < FC) ? b[i] : 0.0f;
}

// One wave (32 lanes) per edge; each lane moves 4 contiguous floats (128 total).
__global__ void agg_kernel(const float* __restrict__ x, const int* __restrict__ src,
                           const int* __restrict__ dst, float* __restrict__ sum, int E) {
  int t = blockIdx.x * blockDim.x + threadIdx.x;
  int e = t >> 5;
  if (e >= E) return;
  int lane = t & 31;
  int s = src[e];
  int d = dst[e];
  const float4 v = *reinterpret_cast<const float4*>(x + (size_t)s * FH + lane * 4);
  float* o = sum + (size_t)d * FH + lane * 4;
  atomicAdd(o + 0, v.x);
  atomicAdd(o + 1, v.y);
  atomicAdd(o + 2, v.z);
  atomicAdd(o + 3, v.w);
}

// Fused SAGE layer GEMM: out = act(A_self @ W_self + (A_neigh * inv_deg) @ W_neigh + b)
// One wave per 16x16 output tile. grid = (M/16, NOUT/16), block = 32.
// NOUT is the padded weight width (multiple of 16); NOUT_REAL is the true
// output width / leading dimension of `out`.
//
// fp32 WMMA 16x16x4 fragment layout (ISA 7.12.2):
//   A 16x4 : lane -> M = lane&15 ; VGPR{0,1} hold K = kk + 2*(lane>>4) + {0,1}
//   B 4x16 : lane -> N = lane&15 ; VGPR{0,1} hold K = kk + 2*(lane>>4) + {0,1}
//   C/D    : lane -> N = lane&15 ; VGPR r -> M = r + 8*(lane>>4)
template <int K, int NOUT, int NOUT_REAL, bool RELU>
__global__ void sage_gemm(const float* __restrict__ A_self,
                          const float* __restrict__ A_neigh,
                          const float* __restrict__ inv_deg,
                          const float* __restrict__ W_self,
                          const float* __restrict__ W_neigh,
                          const float* __restrict__ bias,
                          float* __restrict__ out) {
  // Row stride K+4 = 132 floats: 528 B rows are 16B-aligned (float4 stores OK)
  // and advance 4 banks per row -> the 16 lanes of a b64 fragment read hit
  // 32 distinct banks (no conflicts).
  constexpr int LDSS = K + 4;
  __shared__ float As[16 * LDSS];
  __shared__ float An[16 * LDSS];

  const int lane = threadIdx.x;      // blockDim.x == 32 (wave32)
  const int l16  = lane & 15;
  const int half = lane >> 4;
  const int m_base = blockIdx.x * 16;
  const int n_base = blockIdx.y * 16;

  // Cooperative, coalesced staging of both 16xK A tiles (neigh pre-scaled).
  constexpr int ROW_F4 = K / 4;                    // float4s per row (32)
#pragma unroll
  for (int i = 0; i < (16 * ROW_F4) / 32; ++i) {   // 16 iterations for K=128
    const int f4 = i * 32 + lane;
    const int r  = f4 / ROW_F4;
    const int c4 = f4 % ROW_F4;
    const size_t goff = (size_t)(m_base + r) * K + c4 * 4;
    const float4 vs = *reinterpret_cast<const float4*>(A_self + goff);
    float4 vn       = *reinterpret_cast<const float4*>(A_neigh + goff);
    const float sc  = inv_deg[m_base + r];
    vn.x *= sc; vn.y *= sc; vn.z *= sc; vn.w *= sc;
    *reinterpret_cast<float4*>(&As[r * LDSS + c4 * 4]) = vs;
    *reinterpret_cast<float4*>(&An[r * LDSS + c4 * 4]) = vn;
  }
  __syncthreads();

  const int col = n_base + l16;
  const float* __restrict__ as = &As[l16 * LDSS];
  const float* __restrict__ an = &An[l16 * LDSS];

  v8f acc = {};
#pragma unroll 8
  for (int kk = 0; kk < K; kk += 4) {
    const int k0 = kk + 2 * half;
    // self contribution
    v2f a = *reinterpret_cast<const v2f*>(as + k0);        // ds_load_b64
    v2f b;
    b.x = W_self[(size_t)k0 * NOUT + col];                 // coalesced, L2-hot
    b.y = W_self[(size_t)(k0 + 1) * NOUT + col];
    acc = __builtin_amdgcn_wmma_f32_16x16x4_f32(false, a, false, b,
                                                (short)0, acc, false, false);
    // neighbor contribution (inv_deg already folded in during staging)
    a = *reinterpret_cast<const v2f*>(an + k0);
    b.x = W_neigh[(size_t)k0 * NOUT + col];
    b.y = W_neigh[(size_t)(k0 + 1) * NOUT + col];
    acc = __builtin_amdgcn_wmma_f32_16x16x4_f32(false, a, false, b,
                                                (short)0, acc, false, false);
  }

  if ((NOUT_REAL % 16) != 0 && col >= NOUT_REAL) return;   // post-WMMA: safe
  const float bv = bias[col];
#pragma unroll
  for (int r = 0; r < 8; ++r) {
    const int m = m_base + r + 8 * half;
    float v = acc[r] + bv;
    if (RELU) v = fmaxf(v, 0.0f);
    out[(size_t)m * NOUT_REAL + col] = v;
  }
}

extern "C" void kernel_launch(void* const* d_in, const int* in_sizes, int n_in,
                              void* d_out, int out_size, void* d_ws, size_t ws_size,
                              hipStream_t stream) {
  const float* x   = (const float*)d_in[0];
  const int*   src = (const int*)d_in[1];
  const int*   dst = (const int*)d_in[2];
  const float* Ws1 = (const float*)d_in[3];
  const float* Wn1 = (const float*)d_in[4];
  const float* b1  = (const float*)d_in[5];
  const float* Ws2 = (const float*)d_in[6];
  const float* Wn2 = (const float*)d_in[7];
  const float* b2  = (const float*)d_in[8];
  const float* Ws3 = (const float*)d_in[9];
  const float* Wn3 = (const float*)d_in[10];
  const float* b3  = (const float*)d_in[11];
  float* out = (float*)d_out;

  // Workspace layout (floats):
  //   inv_deg[NN] | agg[NN*FH] | h1[NN*FH] | h2[NN*FH] | Ws3p[FH*FCP] | Wn3p[FH*FCP] | b3p[FCP]
  float* inv  = (float*)d_ws;
  float* agg  = inv  + NN;
  float* h1   = agg  + (size_t)NN * FH;
  float* h2   = h1   + (size_t)NN * FH;
  float* Ws3p = h2   + (size_t)NN * FH;
  float* Wn3p = Ws3p + FH * FCP;
  float* b3p  = Wn3p + FH * FCP;

  const int TB = 256;
  const int gb_node = (NN + TB - 1) / TB;
  const int gb_feat = (int)(((size_t)NN * FH + TB - 1) / TB);
  const int gb_edge = (NE + TB - 1) / TB;
  const int gb_agg  = (int)(((size_t)NE * 32 + TB - 1) / TB);   // 32 lanes/edge
  const int gb_pack = (FH * FCP + TB - 1) / TB;

  const dim3 gemm_block(32, 1, 1);
  const dim3 gemm_grid_h(NN / 16, FH / 16, 1);      // 3125 x 8
  const dim3 gemm_grid_c(NN / 16, FCP / 16, 1);     // 3125 x 3

  // degree -> inv_deg (reused by all layers); pad layer-3 weights once
  zero_f32<<<gb_node, TB, 0, stream>>>(inv, NN);
  degree_kernel<<<gb_edge, TB, 0, stream>>>(dst, inv, NE);
  invdeg_kernel<<<gb_node, TB, 0, stream>>>(inv, NN);
  pack_pad<<<gb_pack, TB, 0, stream>>>(Ws3, Wn3, b3, Ws3p, Wn3p, b3p);

  // layer 1: h1 = relu(x@Ws1 + mean_agg(x)@Wn1 + b1)
  zero_f32<<<gb_feat, TB, 0, stream>>>(agg, NN * FH);
  agg_kernel<<<gb_agg, TB, 0, stream>>>(x, src, dst, agg, NE);
  sage_gemm<FIN, FH, FH, true><<<gemm_grid_h, gemm_block, 0, stream>>>(
      x, agg, inv, Ws1, Wn1, b1, h1);

  // layer 2: h2 = relu(h1@Ws2 + mean_agg(h1)@Wn2 + b2)
  zero_f32<<<gb_feat, TB, 0, stream>>>(agg, NN * FH);
  agg_kernel<<<gb_agg, TB, 0, stream>>>(h1, src, dst, agg, NE);
  sage_gemm<FH, FH, FH, true><<<gemm_grid_h, gemm_block, 0, stream>>>(
      h1, agg, inv, Ws2, Wn2, b2, h2);

  // layer 3: out = h2@Ws3 + mean_agg(h2)@Wn3 + b3   (no relu, 47 real cols)
  zero_f32<<<gb_feat, TB, 0, stream>>>(agg, NN * FH);
  agg_kernel<<<gb_agg, TB, 0, stream>>>(h2, src, dst, agg, NE);
  sage_gemm<FH, FCP, FC, false><<<gemm_grid_c, gemm_block, 0, stream>>>(
      h2, agg, inv, Ws3p, Wn3p, b3p, out);
}